// linear_development_10746008175153
// MI455X (gfx1250) — compile-verified
//
#include <hip/hip_runtime.h>
#include <hip/hip_bf16.h>

// ---------------------------------------------------------------------------
// CDNA5 (gfx1250) path-development kernel.
// Core op: 16x16 f32 matmul = 4x chained V_WMMA_F32_16X16X4_F32 (wave32 tile).
// ---------------------------------------------------------------------------

typedef float v2f __attribute__((ext_vector_type(2)));
typedef float v8f __attribute__((ext_vector_type(8)));

#define DEV_INLINE __device__ __forceinline__

static constexpr int B_      = 64;
static constexpr int T_      = 2048;
static constexpr int CIN_    = 8;
static constexpr int M_      = 16;
static constexpr int O_      = 10;
static constexpr int TSTEPS  = T_ - 1;          // 2047
static constexpr int CHUNKS  = 32;              // time chunks per batch
static constexpr int SPC     = 64;              // steps per chunk (32*64 >= 2047)

// Explicit LDS dependency fence (CDNA5 split counters). In-wave DS ops are
// issued in order; waiting DScnt==0 makes all lanes' LDS writes visible to
// subsequent in-wave LDS reads without a workgroup barrier.
DEV_INLINE void lds_fence() {
  asm volatile("s_wait_dscnt 0" ::: "memory");
}

// ---- WMMA f32 16x16 tile helpers (wave32) ---------------------------------
// C/D layout: VGPR r -> element (row = r + (lane>=16 ? 8 : 0), col = lane&15).

DEV_INLINE void store_C_to_lds(float* buf, const v8f& m, int lane) {
  const int col   = lane & 15;
  const int rbase = (lane >> 4) << 3;   // 0 or 8
#pragma unroll
  for (int r = 0; r < 8; ++r) buf[(rbase + r) * 16 + col] = m[r];
}

DEV_INLINE void load_C_from_global(const float* src, v8f& m, int lane) {
  const int col   = lane & 15;
  const int rbase = (lane >> 4) << 3;
#pragma unroll
  for (int r = 0; r < 8; ++r) m[r] = src[(rbase + r) * 16 + col];
}

DEV_INLINE void store_C_to_global(float* dst, const v8f& m, int lane) {
  const int col   = lane & 15;
  const int rbase = (lane >> 4) << 3;
#pragma unroll
  for (int r = 0; r < 8; ++r) dst[(rbase + r) * 16 + col] = m[r];
}

// A-operand layout for 16x4 f32 slice kk (K = 4*kk .. 4*kk+3):
// lanes 0-15: M=lane,   VGPR0=K0, VGPR1=K1 ; lanes 16-31: M=lane-16, K2,K3.
DEV_INLINE void load_A_from_lds(const float* buf, v2f a[4], int lane) {
  const int mrow = lane & 15;
  const int koff = (lane >> 4) << 1;    // 0 or 2
#pragma unroll
  for (int kk = 0; kk < 4; ++kk) {
    a[kk].x = buf[mrow * 16 + kk * 4 + koff + 0];
    a[kk].y = buf[mrow * 16 + kk * 4 + koff + 1];
  }
}

// B-operand layout for 4x16 f32 slice kk ("row striped across lanes"):
// VGPR0: lanes 0-15 -> K=4kk+0, lanes 16-31 -> K=4kk+2; VGPR1: +1 / +3.
DEV_INLINE void load_B_from_lds(const float* buf, v2f b[4], int lane) {
  const int ncol = lane & 15;
  const int koff = (lane >> 4) << 1;    // 0 or 2
#pragma unroll
  for (int kk = 0; kk < 4; ++kk) {
    b[kk].x = buf[(kk * 4 + koff + 0) * 16 + ncol];
    b[kk].y = buf[(kk * 4 + koff + 1) * 16 + ncol];
  }
}

// Full 16x16x16 f32 product via 4 chained K=4 WMMAs.
DEV_INLINE v8f wmma_mac4(const v2f a[4], const v2f b[4]) {
  v8f acc = {};
#pragma unroll
  for (int kk = 0; kk < 4; ++kk) {
    acc = __builtin_amdgcn_wmma_f32_16x16x4_f32(
        /*neg_a=*/false, a[kk], /*neg_b=*/false, b[kk],
        /*c_mod=*/(short)0, acc, /*reuse_a=*/false, /*reuse_b=*/false);
  }
  return acc;
}

DEV_INLINE v8f identity_C(int lane) {
  const int col   = lane & 15;
  const int rbase = (lane >> 4) << 3;
  v8f I;
#pragma unroll
  for (int r = 0; r < 8; ++r) I[r] = (rbase + r == col) ? 1.0f : 0.0f;
  return I;
}

// ---------------------------------------------------------------------------
// Kernel 1: per-wave chunk of time steps. S -> expm(S) -> running product.
// ---------------------------------------------------------------------------
__global__ __launch_bounds__(256)
void dev_chunk_kernel(const float* __restrict__ X,
                      const float* __restrict__ A,
                      float* __restrict__ chunkP) {
  const int lane  = threadIdx.x & 31;
  const int wib   = threadIdx.x >> 5;                 // wave in block (0..7)
  const int gwave = blockIdx.x * 8 + wib;             // 0..2047
  const int b     = gwave / CHUNKS;
  const int chunk = gwave % CHUNKS;

  __shared__ float lds[8 * 2 * 256];
  float* bufA = &lds[(wib * 2 + 0) * 256];
  float* bufB = &lds[(wib * 2 + 1) * 256];

  const int col   = lane & 15;
  const int rbase = (lane >> 4) << 3;

  // Skew generators G_c in C/D layout: Greg[c][r] = A[row][col][c]-A[col][row][c]
  float Greg[CIN_][8];
#pragma unroll
  for (int r = 0; r < 8; ++r) {
    const int row = rbase + r;
#pragma unroll
    for (int c = 0; c < CIN_; ++c) {
      const float aij = A[row * (M_ * CIN_) + col * CIN_ + c];
      const float aji = A[col * (M_ * CIN_) + row * CIN_ + c];
      Greg[c][r] = aij - aji;
    }
  }

  v8f P = identity_C(lane);

  const int t0 = chunk * SPC;
  const int t1 = (t0 + SPC < TSTEPS) ? (t0 + SPC) : TSTEPS;
  const float* Xb = X + (size_t)b * T_ * CIN_;

  // carry x(t) across iterations
  float4 xp0 = *(const float4*)(Xb + (size_t)t0 * CIN_);
  float4 xp1 = *(const float4*)(Xb + (size_t)t0 * CIN_ + 4);

  for (int t = t0; t < t1; ++t) {
    const float4 xc0 = *(const float4*)(Xb + (size_t)(t + 1) * CIN_);
    const float4 xc1 = *(const float4*)(Xb + (size_t)(t + 1) * CIN_ + 4);
    float dX[CIN_];
    dX[0] = xc0.x - xp0.x; dX[1] = xc0.y - xp0.y;
    dX[2] = xc0.z - xp0.z; dX[3] = xc0.w - xp0.w;
    dX[4] = xc1.x - xp1.x; dX[5] = xc1.y - xp1.y;
    dX[6] = xc1.z - xp1.z; dX[7] = xc1.w - xp1.w;
    xp0 = xc0; xp1 = xc1;

    // S (scaled by 2^-6) in C/D layout
    v8f Ss;
#pragma unroll
    for (int r = 0; r < 8; ++r) {
      float s = 0.0f;
#pragma unroll
      for (int c = 0; c < CIN_; ++c) s = fmaf(dX[c], Greg[c][r], s);
      Ss[r] = s * 0.015625f;  // 1/64
    }

    // A-layout of Ss cached for the whole Taylor recurrence.
    store_C_to_lds(bufA, Ss, lane);
    lds_fence();
    v2f aS[4];
    load_A_from_lds(bufA, aS, lane);

    // Taylor: T = I + Ss + Ss^2/2! + ... + Ss^8/8!  (term recurrence)
    v8f Tm = Ss;
#pragma unroll
    for (int r = 0; r < 8; ++r) Tm[r] += (rbase + r == col) ? 1.0f : 0.0f;
    v8f term = Ss;
#pragma unroll
    for (int k = 2; k <= 8; ++k) {
      store_C_to_lds(bufB, term, lane);
      lds_fence();
      v2f bT[4];
      load_B_from_lds(bufB, bT, lane);
      v8f prod = wmma_mac4(aS, bT);
      const float inv = 1.0f / (float)k;
#pragma unroll
      for (int r = 0; r < 8; ++r) {
        term[r] = prod[r] * inv;
        Tm[r]  += term[r];
      }
    }

    // Undo scaling: square 6 times.
#pragma unroll
    for (int sq = 0; sq < 6; ++sq) {
      store_C_to_lds(bufA, Tm, lane);
      lds_fence();
      v2f aT[4], bT[4];
      load_A_from_lds(bufA, aT, lane);
      load_B_from_lds(bufA, bT, lane);
      Tm = wmma_mac4(aT, bT);
    }

    // Running product: P = expm(S_t) @ P   (later factor on the left)
    store_C_to_lds(bufA, Tm, lane);
    store_C_to_lds(bufB, P, lane);
    lds_fence();
    v2f aM[4], bP[4];
    load_A_from_lds(bufA, aM, lane);
    load_B_from_lds(bufB, bP, lane);
    P = wmma_mac4(aM, bP);
  }

  store_C_to_global(chunkP + (size_t)(b * CHUNKS + chunk) * 256, P, lane);
}

// ---------------------------------------------------------------------------
// Kernel 2: per-batch reduction of chunk products (time order, left-multiply).
// One wave per block.
// ---------------------------------------------------------------------------
__global__ __launch_bounds__(32)
void dev_reduce_kernel(const float* __restrict__ chunkP,
                       float* __restrict__ Z) {
  const int b    = blockIdx.x;
  const int lane = threadIdx.x & 31;

  __shared__ float bufA[256];
  __shared__ float bufB[256];

  v8f P = identity_C(lane);
  for (int c = 0; c < CHUNKS; ++c) {
    v8f Mc;
    load_C_from_global(chunkP + (size_t)(b * CHUNKS + c) * 256, Mc, lane);
    store_C_to_lds(bufA, Mc, lane);
    store_C_to_lds(bufB, P, lane);
    lds_fence();
    v2f aM[4], bP[4];
    load_A_from_lds(bufA, aM, lane);
    load_B_from_lds(bufB, bP, lane);
    P = wmma_mac4(aM, bP);
  }
  store_C_to_global(Z + (size_t)b * 256, P, lane);
}

// ---------------------------------------------------------------------------
// Kernel 3: out[b,o] = sum_k Z[b,k] * W[o,k] + bias[o]   (64x10, trivial)
// ---------------------------------------------------------------------------
__global__ __launch_bounds__(256)
void dev_head_kernel(const float* __restrict__ Z,
                     const float* __restrict__ W,
                     const float* __restrict__ bias,
                     float* __restrict__ out) {
  const int idx = blockIdx.x * blockDim.x + threadIdx.x;
  if (idx >= B_ * O_) return;
  const int b = idx / O_;
  const int o = idx % O_;
  float acc = bias[o];
  const float* z = Z + (size_t)b * 256;
  const float* w = W + (size_t)o * 256;
#pragma unroll 8
  for (int k = 0; k < 256; ++k) acc = fmaf(z[k], w[k], acc);
  out[idx] = acc;
}

// ---------------------------------------------------------------------------
extern "C" void kernel_launch(void* const* d_in, const int* in_sizes, int n_in,
                              void* d_out, int out_size, void* d_ws, size_t ws_size,
                              hipStream_t stream) {
  const float* X    = (const float*)d_in[0];   // [64,2048,8]
  const float* A    = (const float*)d_in[1];   // [16,16,8]
  const float* W    = (const float*)d_in[2];   // [10,256]
  const float* bias = (const float*)d_in[3];   // [10]
  float* out = (float*)d_out;                  // [64,10]

  float* chunkP = (float*)d_ws;                                  // 64*32*256 f32 = 2 MB
  float* Z      = (float*)((char*)d_ws + (size_t)B_ * CHUNKS * 256 * sizeof(float));

  // 2048 waves, 8 waves/block -> 256 blocks of 256 threads.
  dev_chunk_kernel<<<(B_ * CHUNKS) / 8, 256, 0, stream>>>(X, A, chunkP);
  dev_reduce_kernel<<<B_, 32, 0, stream>>>(chunkP, Z);
  dev_head_kernel<<<(B_ * O_ + 255) / 256, 256, 0, stream>>>(Z, W, bias, out);
}